// ThresholdTupleMatcher_8839042695246
// MI455X (gfx1250) — compile-verified
//
#include <hip/hip_runtime.h>
#include <hip/hip_bf16.h>
#include <math.h>

typedef __attribute__((ext_vector_type(2))) float v2f;
typedef __attribute__((ext_vector_type(8))) float v8f;

#define B_   256
#define S_   4
#define W1_  64
#define W2_  128
#define D_   300
#define HID_ 4
#define EPS_ 1e-7f

#define KC_  20          // K-chunk staged in LDS (300 = 15 * 20)
#define NCHUNK_ (D_ / KC_)

// One workgroup per (b,s): 8 wave32 waves compute the 64x128 fp32 GEMM with
// V_WMMA_F32_16X16X4_F32 out of LDS-staged K-chunks (coalesced global loads,
// conflict-free ds_load_b64 fragment reads), then threshold+mask+reduce.
__global__ __launch_bounds__(256) void ttm_sim_kernel(
    const float* __restrict__ t1,      // (B,S,W1,D)
    const float* __restrict__ t2,      // (B,S,W2,D)
    const int*   __restrict__ mask1,   // (B,S,W1)
    const int*   __restrict__ mask2,   // (B,S,W2)
    const float* __restrict__ thrPtr,  // scalar
    float*       __restrict__ norm_out // (B*S)
) {
    __shared__ float As[W1_ * KC_];    // 64 x 20, row stride 20 dwords
    __shared__ float Bs[W2_ * KC_];    // 128 x 20
    __shared__ float red[2];           // [0]=overlap sum, [1]=count1

    const int bs = blockIdx.x;  // 0 .. B*S-1
    const float* Abase = t1 + (size_t)bs * W1_ * D_;
    const float* Bbase = t2 + (size_t)bs * W2_ * D_;
    const int*   m1    = mask1 + bs * W1_;
    const int*   m2    = mask2 + bs * W2_;

    const int tid   = threadIdx.x;
    const int wave  = tid >> 5;     // 0..7
    const int lane  = tid & 31;
    const int lmod  = lane & 15;    // 0..15
    const int lhalf = lane >> 4;    // 0 or 1

    const int mBase = (wave & 3) * 16;   // M tile: 0,16,32,48
    const int nBase = (wave >> 2) * 64;  // N strip: 0 or 64 (4 tiles of 16)

    // ---- staging assignment (coalesced): A: 4 threads/row x 5 cols,
    //                                      B: 2 threads/row x 10 cols ----
    const int aRow = tid >> 2;           // 0..63
    const int aCol = (tid & 3) * 5;      // 0,5,10,15
    const int bRow = tid >> 1;           // 0..127
    const int bCol = (tid & 1) * 10;     // 0,10
    const float* aG = Abase + (size_t)aRow * D_ + aCol;
    const float* bG = Bbase + (size_t)bRow * D_ + bCol;
    float* aL = &As[aRow * KC_ + aCol];
    float* bL = &Bs[bRow * KC_ + bCol];

    // ---- WMMA fragment LDS offsets ----
    // A lane holds A[m][k0+2*lhalf], A[m][k0+2*lhalf+1]; m = lane%16.
    // B lane holds B[k][n] = t2[n][k]; n = lane%16.
    const float* aF  = &As[(mBase + lmod) * KC_ + 2 * lhalf];
    const float* bF0 = &Bs[(nBase +  0 + lmod) * KC_ + 2 * lhalf];
    const float* bF1 = &Bs[(nBase + 16 + lmod) * KC_ + 2 * lhalf];
    const float* bF2 = &Bs[(nBase + 32 + lmod) * KC_ + 2 * lhalf];
    const float* bF3 = &Bs[(nBase + 48 + lmod) * KC_ + 2 * lhalf];

    v8f acc0 = {}, acc1 = {}, acc2 = {}, acc3 = {};

    float pa[5], pb[10];
    // prefetch chunk 0
#pragma unroll
    for (int i = 0; i < 5; ++i)  pa[i] = aG[i];
#pragma unroll
    for (int i = 0; i < 10; ++i) pb[i] = bG[i];

    for (int c = 0; c < NCHUNK_; ++c) {
        // commit prefetched chunk to LDS
#pragma unroll
        for (int i = 0; i < 5; ++i)  aL[i] = pa[i];
#pragma unroll
        for (int i = 0; i < 10; ++i) bL[i] = pb[i];
        __syncthreads();

        // issue global loads for next chunk (overlaps with compute below)
        if (c + 1 < NCHUNK_) {
            const int k0 = (c + 1) * KC_;
#pragma unroll
            for (int i = 0; i < 5; ++i)  pa[i] = aG[k0 + i];
#pragma unroll
            for (int i = 0; i < 10; ++i) pb[i] = bG[k0 + i];
        }

        // 5 K-steps of 4 from LDS: conflict-free ds_load_b64 fragments
#pragma unroll
        for (int kk = 0; kk < KC_; kk += 4) {
            v2f a  = *(const v2f*)(aF  + kk);
            v2f b0 = *(const v2f*)(bF0 + kk);
            v2f b1 = *(const v2f*)(bF1 + kk);
            v2f b2 = *(const v2f*)(bF2 + kk);
            v2f b3 = *(const v2f*)(bF3 + kk);
            acc0 = __builtin_amdgcn_wmma_f32_16x16x4_f32(
                false, a, false, b0, (short)0, acc0, false, false);
            acc1 = __builtin_amdgcn_wmma_f32_16x16x4_f32(
                false, a, false, b1, (short)0, acc1, false, false);
            acc2 = __builtin_amdgcn_wmma_f32_16x16x4_f32(
                false, a, false, b2, (short)0, acc2, false, false);
            acc3 = __builtin_amdgcn_wmma_f32_16x16x4_f32(
                false, a, false, b3, (short)0, acc3, false, false);
        }
        __syncthreads();
    }

    // ---- masked threshold count ----
    const float thr = *thrPtr;

    // C/D layout: VGPR v -> m = mBase + v + 8*lhalf ; n = nTileBase + lmod
    float m1v[8];
#pragma unroll
    for (int v = 0; v < 8; ++v)
        m1v[v] = (float)m1[mBase + 8 * lhalf + v];

    const float f20 = (float)m2[nBase +  0 + lmod];
    const float f21 = (float)m2[nBase + 16 + lmod];
    const float f22 = (float)m2[nBase + 32 + lmod];
    const float f23 = (float)m2[nBase + 48 + lmod];

    float s0 = 0.f, s1 = 0.f, s2 = 0.f, s3 = 0.f;
#pragma unroll
    for (int v = 0; v < 8; ++v) {
        s0 += (acc0[v] >= thr ? 1.f : 0.f) * m1v[v];
        s1 += (acc1[v] >= thr ? 1.f : 0.f) * m1v[v];
        s2 += (acc2[v] >= thr ? 1.f : 0.f) * m1v[v];
        s3 += (acc3[v] >= thr ? 1.f : 0.f) * m1v[v];
    }
    float partial = s0 * f20 + s1 * f21 + s2 * f22 + s3 * f23;

    // ---- workgroup reduction via LDS float atomics (ds_add_f32) ----
    if (tid < 2) red[tid] = 0.f;
    __syncthreads();
    atomicAdd(&red[0], partial);
    if (tid < W1_) atomicAdd(&red[1], (float)m1[tid]);
    __syncthreads();

    if (tid == 0) {
        const float c   = red[1];
        const float div = (c == 0.f) ? EPS_ : c;
        norm_out[bs] = red[0] / div;
    }
}

// Tiny head: h = tanh(norm @ W_hidden), out = sigmoid(h @ W_score).
__global__ __launch_bounds__(256) void ttm_head_kernel(
    const float* __restrict__ norm,  // (B,S)
    const float* __restrict__ Wh,    // (S,HID)
    const float* __restrict__ Wsc,   // (HID,1)
    float*       __restrict__ out    // (B,1)
) {
    const int b = blockIdx.x * blockDim.x + threadIdx.x;
    if (b >= B_) return;

    float n[S_];
#pragma unroll
    for (int s = 0; s < S_; ++s) n[s] = norm[b * S_ + s];

    float y = 0.f;
#pragma unroll
    for (int j = 0; j < HID_; ++j) {
        float hj = 0.f;
#pragma unroll
        for (int s = 0; s < S_; ++s) hj += n[s] * Wh[s * HID_ + j];
        y += tanhf(hj) * Wsc[j];
    }
    out[b] = 1.f / (1.f + expf(-y));
}

extern "C" void kernel_launch(void* const* d_in, const int* in_sizes, int n_in,
                              void* d_out, int out_size, void* d_ws, size_t ws_size,
                              hipStream_t stream) {
    (void)in_sizes; (void)n_in; (void)out_size; (void)ws_size;
    const float* t1  = (const float*)d_in[0];
    const float* t2  = (const float*)d_in[1];
    const int*   m1  = (const int*)d_in[2];
    const int*   m2  = (const int*)d_in[3];
    const float* thr = (const float*)d_in[4];
    const float* Wh  = (const float*)d_in[5];
    const float* Wsc = (const float*)d_in[6];

    float* norm = (float*)d_ws;  // B*S floats of scratch

    ttm_sim_kernel<<<B_ * S_, 256, 0, stream>>>(t1, t2, m1, m2, thr, norm);
    ttm_head_kernel<<<1, 256, 0, stream>>>(norm, Wh, Wsc, (float*)d_out);
}